// GATModel_47588237639831
// MI455X (gfx1250) — compile-verified
//
#include <hip/hip_runtime.h>
#include <math.h>
#include <stdint.h>

typedef float v2f __attribute__((ext_vector_type(2)));
typedef float v8f __attribute__((ext_vector_type(8)));

// ---------------------------------------------------------------------------
// FP32 WMMA GEMM: C[M,N] = A[M,K] @ B[K,N] (+ bias[N] if bias != nullptr)
// Block = 128 threads (4 waves), 64x64 C tile, each wave a 16(M)x64(N) strip.
// Global->LDS staging uses CDNA5 async copies (global_load_async_to_lds_b128,
// ASYNCcnt) with double-buffered LDS tiles so the next K-chunk streams in
// while the current one is consumed by v_wmma_f32_16x16x4_f32.
// Requires M%64==0, N%64==0, K%32==0 (true for all GEMMs in this model).
// ---------------------------------------------------------------------------
#define AS_STRIDE 36   // 144 B rows: 16B-aligned, conflict-free A-frag reads
#define BS_STRIDE 72   // 288 B rows: 16B-aligned, half-waves hit disjoint banks

__global__ __launch_bounds__(128) void gemm_f32_wmma(
    const float* __restrict__ A, const float* __restrict__ B,
    float* __restrict__ C, const float* __restrict__ bias,
    int M, int N, int K)
{
  __shared__ float As[2][64][AS_STRIDE];
  __shared__ float Bs[2][32][BS_STRIDE];

  const int tid    = threadIdx.x;
  const int wave   = tid >> 5;
  const int lane   = tid & 31;
  const int lane16 = lane & 15;
  const int half   = lane >> 4;            // 0: K+{0,1} / M+r, 1: K+{2,3} / M+r+8
  const int m0     = blockIdx.y * 64;
  const int n0     = blockIdx.x * 64;

  v8f acc[4];
#pragma unroll
  for (int s = 0; s < 4; ++s)
#pragma unroll
    for (int r = 0; r < 8; ++r) acc[s][r] = 0.0f;

  const int arow = tid >> 3;         // 0..15
  const int acol = (tid & 7) * 4;    // 0..28
  const int brow = tid >> 4;         // 0..7
  const int bcol = (tid & 15) * 4;   // 0..60

  // issue async copies of one 64x32 A tile + 32x64 B tile into LDS buffer
  auto issue_tile = [&](int buf, int k0) {
#pragma unroll
    for (int i = 0; i < 4; ++i) {
      const int r = arow + i * 16;
      const float* g = A + (size_t)(m0 + r) * K + (k0 + acol);
      const unsigned l = (unsigned)(uintptr_t)&As[buf][r][acol];
      asm volatile("global_load_async_to_lds_b128 %0, %1, off"
                   :: "v"(l), "v"(g) : "memory");
    }
#pragma unroll
    for (int i = 0; i < 4; ++i) {
      const int r = brow + i * 8;
      const float* g = B + (size_t)(k0 + r) * N + (n0 + bcol);
      const unsigned l = (unsigned)(uintptr_t)&Bs[buf][r][bcol];
      asm volatile("global_load_async_to_lds_b128 %0, %1, off"
                   :: "v"(l), "v"(g) : "memory");
    }
  };

  issue_tile(0, 0);
  asm volatile("s_wait_asynccnt 0x0" ::: "memory");
  __syncthreads();

  const int nch = K >> 5;
  for (int ch = 0; ch < nch; ++ch) {
    const int buf = ch & 1;
    if (ch + 1 < nch) issue_tile(buf ^ 1, (ch + 1) << 5);  // prefetch next chunk

    const int mrow = wave * 16 + lane16;
#pragma unroll
    for (int kk = 0; kk < 32; kk += 4) {
      // A fragment 16x4: lanes0-15 -> K=kk+0/1, lanes16-31 -> K=kk+2/3
      v2f a;
      a.x = As[buf][mrow][kk + 2 * half + 0];
      a.y = As[buf][mrow][kk + 2 * half + 1];
#pragma unroll
      for (int s = 0; s < 4; ++s) {
        // B fragment 4x16, mirrored layout
        v2f b;
        b.x = Bs[buf][kk + 2 * half + 0][s * 16 + lane16];
        b.y = Bs[buf][kk + 2 * half + 1][s * 16 + lane16];
        acc[s] = __builtin_amdgcn_wmma_f32_16x16x4_f32(
            false, a, false, b, (short)0, acc[s], false, false);
      }
    }
    // own async copies done, then all waves' copies visible / reads finished
    asm volatile("s_wait_asynccnt 0x0" ::: "memory");
    __syncthreads();
  }

  // C/D layout: VGPR r -> M = r (lanes0-15) / r+8 (lanes16-31), N = lane16
#pragma unroll
  for (int s = 0; s < 4; ++s) {
    const int col = n0 + s * 16 + lane16;
    const float badd = bias ? bias[col] : 0.0f;
#pragma unroll
    for (int r = 0; r < 8; ++r) {
      const int row = m0 + wave * 16 + half * 8 + r;
      C[(size_t)row * N + col] = acc[s][r] + badd;
    }
  }
}

// ---------------------------------------------------------------------------
// Per (node,head) attention coefficients: a_s/a_d = <h[n,h,:], att_*[h,:]>
// One wave per (n,h); butterfly-shuffle reduction.
// ---------------------------------------------------------------------------
__global__ __launch_bounds__(256) void att_dots(
    const float* __restrict__ h, const float* __restrict__ w_s,
    const float* __restrict__ w_d, float* __restrict__ a_s,
    float* __restrict__ a_d, int NH, int H, int C)
{
  const int wid  = (int)(((size_t)blockIdx.x * blockDim.x + threadIdx.x) >> 5);
  const int lane = threadIdx.x & 31;
  if (wid >= NH) return;
  const int hh = wid % H;
  const float* hp = h   + (size_t)wid * C;
  const float* sp = w_s + (size_t)hh * C;
  const float* dp = w_d + (size_t)hh * C;
  float s = 0.f, d = 0.f;
  for (int c = lane; c < C; c += 32) {
    const float v = hp[c];
    s += v * sp[c];
    d += v * dp[c];
  }
  for (int o = 16; o > 0; o >>= 1) {
    s += __shfl_xor(s, o, 32);
    d += __shfl_xor(d, o, 32);
  }
  if (lane == 0) { a_s[wid] = s; a_d[wid] = d; }
}

// order-preserving float<->uint keys for atomic segment-max
__device__ __forceinline__ unsigned f32_key(float f) {
  const unsigned u = __float_as_uint(f);
  return (u & 0x80000000u) ? ~u : (u | 0x80000000u);
}
__device__ __forceinline__ float key_f32(unsigned k) {
  const unsigned b = (k & 0x80000000u) ? (k & 0x7fffffffu) : ~k;
  return __uint_as_float(b);
}

__global__ void fill_u32(unsigned* __restrict__ p, unsigned v, size_t n) {
  const size_t i = (size_t)blockIdx.x * blockDim.x + threadIdx.x;
  if (i < n) p[i] = v;
}

// pass 1: e = leaky_relu(a_s[src]+a_d[dst]); segment max over dst
__global__ void edge_logits(
    const int* __restrict__ src, const int* __restrict__ dst,
    const float* __restrict__ a_s, const float* __restrict__ a_d,
    float* __restrict__ ev, unsigned* __restrict__ mx,
    int Ereal, int EN, int H)
{
  const int idx = blockIdx.x * blockDim.x + threadIdx.x;
  if (idx >= EN * H) return;
  const int e  = idx / H;
  const int hh = idx - e * H;
  const int s  = (e < Ereal) ? src[e] : (e - Ereal);   // self-loops appended
  const int d  = (e < Ereal) ? dst[e] : (e - Ereal);
  float v = a_s[s * H + hh] + a_d[d * H + hh];
  v = (v > 0.f) ? v : 0.2f * v;                        // leaky_relu, slope 0.2
  ev[idx] = v;
  atomicMax(&mx[d * H + hh], f32_key(v));
}

// pass 2: p = exp(e - max[dst]); segment sum over dst (p stored in-place)
__global__ void edge_exp(
    const int* __restrict__ dst, float* __restrict__ ev,
    const unsigned* __restrict__ mx, float* __restrict__ sm,
    int Ereal, int EN, int H)
{
  const int idx = blockIdx.x * blockDim.x + threadIdx.x;
  if (idx >= EN * H) return;
  const int e  = idx / H;
  const int hh = idx - e * H;
  const int d  = (e < Ereal) ? dst[e] : (e - Ereal);
  const float p = expf(ev[idx] - key_f32(mx[d * H + hh]));
  ev[idx] = p;
  atomicAdd(&sm[d * H + hh], p);
}

// pass 3: out[dst] += alpha * h[src]; one wave per (edge, head)
__global__ __launch_bounds__(256) void scatter_msg(
    const int* __restrict__ src, const int* __restrict__ dst,
    const float* __restrict__ h, const float* __restrict__ p,
    const float* __restrict__ sm, float* __restrict__ out,
    int Ereal, int EN, int H, int C)
{
  const int gw   = (int)(((size_t)blockIdx.x * blockDim.x + threadIdx.x) >> 5);
  const int lane = threadIdx.x & 31;
  if (gw >= EN * H) return;
  const int e  = gw / H;
  const int hh = gw - e * H;
  const int s  = (e < Ereal) ? src[e] : (e - Ereal);
  const int d  = (e < Ereal) ? dst[e] : (e - Ereal);
  const float alpha = p[gw] / (sm[d * H + hh] + 1e-16f);
  const float* hp = h   + ((size_t)s * H + hh) * C;
  float*       op = out + ((size_t)d * H + hh) * C;
  for (int c = lane; c < C; c += 32)
    atomicAdd(&op[c], hp[c] * alpha);
}

// out = (elu?)(in + bias[j % F])
__global__ void elu_bias(
    const float* __restrict__ in, const float* __restrict__ bias,
    float* __restrict__ outp, size_t n, int F, int do_elu)
{
  const size_t i = (size_t)blockIdx.x * blockDim.x + threadIdx.x;
  if (i >= n) return;
  float v = in[i] + bias[i % F];
  if (do_elu) v = (v > 0.f) ? v : (expf(v) - 1.f);
  outp[i] = v;
}

// concat[m] = [x_graph[mask[m]], x_raw[mask[m]]]
__global__ void gather_concat(
    const float* __restrict__ xg, const float* __restrict__ xe,
    const int* __restrict__ mask, float* __restrict__ cc, int Mrows, int F)
{
  const int idx = blockIdx.x * blockDim.x + threadIdx.x;
  if (idx >= Mrows * F) return;
  const int m = idx / F;
  const int j = idx - m * F;
  const int n = mask[m];
  cc[(size_t)m * 2 * F + j]     = xg[(size_t)n * F + j];
  cc[(size_t)m * 2 * F + F + j] = xe[(size_t)n * F + j];
}

// final 768 -> 2 classifier, one wave per row
__global__ __launch_bounds__(256) void classifier2(
    const float* __restrict__ fcout, const float* __restrict__ cls_w,
    const float* __restrict__ cls_b, float* __restrict__ out, int Mrows, int F)
{
  const int m    = (int)(((size_t)blockIdx.x * blockDim.x + threadIdx.x) >> 5);
  const int lane = threadIdx.x & 31;
  if (m >= Mrows) return;
  float a0 = 0.f, a1 = 0.f;
  const float* rp = fcout + (size_t)m * F;
  for (int c = lane; c < F; c += 32) {
    const float v = rp[c];
    a0 += v * cls_w[c * 2 + 0];
    a1 += v * cls_w[c * 2 + 1];
  }
  for (int o = 16; o > 0; o >>= 1) {
    a0 += __shfl_xor(a0, o, 32);
    a1 += __shfl_xor(a1, o, 32);
  }
  if (lane == 0) {
    out[(size_t)m * 2 + 0] = a0 + cls_b[0];
    out[(size_t)m * 2 + 1] = a1 + cls_b[1];
  }
}

// ---------------------------------------------------------------------------
extern "C" void kernel_launch(void* const* d_in, const int* in_sizes, int n_in,
                              void* d_out, int out_size, void* d_ws, size_t ws_size,
                              hipStream_t stream) {
  (void)in_sizes; (void)n_in; (void)out_size; (void)ws_size;

  constexpr int NN = 8192, EE = 32768, MM = 1024;
  constexpr int DIN = 768, HID = 768, H1 = 8;
  constexpr int EN = EE + NN;            // with self-loops
  constexpr int F1 = H1 * HID;           // 6144
  constexpr int F2 = 1 * HID;            // 768

  const float* x0     = (const float*)d_in[0];
  const int*   ei     = (const int*)d_in[1];
  const int*   src    = ei;
  const int*   dst    = ei + EE;
  const int*   mask   = (const int*)d_in[2];
  const float* W1     = (const float*)d_in[3];
  const float* att_s1 = (const float*)d_in[4];
  const float* att_d1 = (const float*)d_in[5];
  const float* b1     = (const float*)d_in[6];
  const float* W2     = (const float*)d_in[7];
  const float* att_s2 = (const float*)d_in[8];
  const float* att_d2 = (const float*)d_in[9];
  const float* b2     = (const float*)d_in[10];
  const float* fc_w   = (const float*)d_in[11];
  const float* fc_b   = (const float*)d_in[12];
  const float* cls_w  = (const float*)d_in[13];
  const float* cls_b  = (const float*)d_in[14];
  float* out = (float*)d_out;

  // workspace carve-up (floats)
  float* ws = (float*)d_ws;
  size_t o = 0;
  float* h1  = ws + o; o += (size_t)NN * F1;
  float* g1  = ws + o; o += (size_t)NN * F1;
  float* h2  = ws + o; o += (size_t)NN * F2;
  float* g2  = ws + o; o += (size_t)NN * F2;
  float* as1 = ws + o; o += (size_t)NN * H1;
  float* ad1 = ws + o; o += (size_t)NN * H1;
  unsigned* mx1 = (unsigned*)(ws + o); o += (size_t)NN * H1;
  float* sm1 = ws + o; o += (size_t)NN * H1;
  float* ev1 = ws + o; o += (size_t)EN * H1;
  float* as2 = ws + o; o += (size_t)NN;
  float* ad2 = ws + o; o += (size_t)NN;
  unsigned* mx2 = (unsigned*)(ws + o); o += (size_t)NN;
  float* sm2 = ws + o; o += (size_t)NN;
  float* ev2 = ws + o; o += (size_t)EN;
  float* cc  = ws + o; o += (size_t)MM * 2 * HID;
  float* fo  = ws + o; o += (size_t)MM * HID;

  auto cdiv = [](size_t n, size_t b) { return (unsigned)((n + b - 1) / b); };

  // ---- layer 1 -------------------------------------------------------------
  gemm_f32_wmma<<<dim3(F1 / 64, NN / 64), 128, 0, stream>>>(
      x0, W1, h1, nullptr, NN, F1, DIN);

  att_dots<<<cdiv((size_t)NN * H1 * 32, 256), 256, 0, stream>>>(
      h1, att_s1, att_d1, as1, ad1, NN * H1, H1, HID);

  fill_u32<<<cdiv((size_t)NN * H1, 256), 256, 0, stream>>>(mx1, 0u, (size_t)NN * H1);
  fill_u32<<<cdiv((size_t)NN * H1, 256), 256, 0, stream>>>((unsigned*)sm1, 0u, (size_t)NN * H1);
  fill_u32<<<cdiv((size_t)NN * F1, 256), 256, 0, stream>>>((unsigned*)g1, 0u, (size_t)NN * F1);

  edge_logits<<<cdiv((size_t)EN * H1, 256), 256, 0, stream>>>(
      src, dst, as1, ad1, ev1, mx1, EE, EN, H1);
  edge_exp<<<cdiv((size_t)EN * H1, 256), 256, 0, stream>>>(
      dst, ev1, mx1, sm1, EE, EN, H1);
  scatter_msg<<<cdiv((size_t)EN * H1 * 32, 256), 256, 0, stream>>>(
      src, dst, h1, ev1, sm1, g1, EE, EN, H1, HID);

  elu_bias<<<cdiv((size_t)NN * F1, 256), 256, 0, stream>>>(
      g1, b1, h1, (size_t)NN * F1, F1, 1);          // x1 = elu(gat1 + b1), in h1

  // ---- layer 2 (H=1) -------------------------------------------------------
  gemm_f32_wmma<<<dim3(F2 / 64, NN / 64), 128, 0, stream>>>(
      h1, W2, h2, nullptr, NN, F2, F1);

  att_dots<<<cdiv((size_t)NN * 32, 256), 256, 0, stream>>>(
      h2, att_s2, att_d2, as2, ad2, NN, 1, HID);

  fill_u32<<<cdiv((size_t)NN, 256), 256, 0, stream>>>(mx2, 0u, (size_t)NN);
  fill_u32<<<cdiv((size_t)NN, 256), 256, 0, stream>>>((unsigned*)sm2, 0u, (size_t)NN);
  fill_u32<<<cdiv((size_t)NN * F2, 256), 256, 0, stream>>>((unsigned*)g2, 0u, (size_t)NN * F2);

  edge_logits<<<cdiv((size_t)EN, 256), 256, 0, stream>>>(
      src, dst, as2, ad2, ev2, mx2, EE, EN, 1);
  edge_exp<<<cdiv((size_t)EN, 256), 256, 0, stream>>>(
      dst, ev2, mx2, sm2, EE, EN, 1);
  scatter_msg<<<cdiv((size_t)EN * 32, 256), 256, 0, stream>>>(
      src, dst, h2, ev2, sm2, g2, EE, EN, 1, HID);

  elu_bias<<<cdiv((size_t)NN * F2, 256), 256, 0, stream>>>(
      g2, b2, h2, (size_t)NN * F2, F2, 0);          // x2 = gat2 + b2, in h2

  // ---- head ----------------------------------------------------------------
  gather_concat<<<cdiv((size_t)MM * HID, 256), 256, 0, stream>>>(
      h2, x0, mask, cc, MM, HID);

  gemm_f32_wmma<<<dim3(HID / 64, MM / 64), 128, 0, stream>>>(
      cc, fc_w, fo, fc_b, MM, HID, 2 * HID);

  classifier2<<<cdiv((size_t)MM * 32, 256), 256, 0, stream>>>(
      fo, cls_w, cls_b, out, MM, HID);
}